// PlaneCalib_83442624627150
// MI455X (gfx1250) — compile-verified
//
#include <hip/hip_runtime.h>
#include <math.h>

#define BS    8
#define MPTS  500000
#define KEEP  475000          // int(0.95 * 500000)
#define HBINS 65536
#define BPB1  64              // blocks per batch, transform kernel
#define BPB2  16              // blocks per batch, histogram kernels
#define BPB3  16              // blocks per batch, sum kernel
#define P_OFF 8388608ull      // byte offset of cached p[] in workspace

typedef float v2f __attribute__((ext_vector_type(2)));
typedef float v8f __attribute__((ext_vector_type(8)));

// ---------------------------------------------------------------- utilities

__global__ void k_zero(unsigned int* __restrict__ dst, int nwords) {
  int i = blockIdx.x * blockDim.x + threadIdx.x;
  if (i < nwords) dst[i] = 0u;
}

// Rodrigues: R = I + f1*K + f2*K@K  (matches reference eps exactly)
__global__ void k_rt(const float* __restrict__ w, const float* __restrict__ tr,
                     float* __restrict__ Rt) {
  if (threadIdx.x != 0 || blockIdx.x != 0) return;
  float w0 = w[0], w1 = w[1], w2 = w[2];
  float t2 = w0 * w0 + w1 * w1 + w2 * w2;
  float th = sqrtf(t2 + 1e-12f);
  float f1 = sinf(th) / th;
  float f2 = (1.0f - cosf(th)) / (th * th);
  float K[9] = {0.f, -w2, w1, w2, 0.f, -w0, -w1, w0, 0.f};
  float K2[9];
  for (int i = 0; i < 3; ++i)
    for (int j = 0; j < 3; ++j) {
      float s = 0.f;
      for (int k = 0; k < 3; ++k) s += K[i * 3 + k] * K[k * 3 + j];
      K2[i * 3 + j] = s;
    }
  for (int i = 0; i < 3; ++i)
    for (int j = 0; j < 3; ++j)
      Rt[i * 3 + j] = (i == j ? 1.0f : 0.0f) + f1 * K[i * 3 + j] + f2 * K2[i * 3 + j];
  Rt[9] = tr[0]; Rt[10] = tr[1]; Rt[11] = tr[2];
}

// recompute p for point idx = b*M + m  (used when ws is too small to cache p)
__device__ __forceinline__ void point_p(const float* __restrict__ x,
                                        const float* __restrict__ y,
                                        const float* rt, size_t idx,
                                        float& p0, float& p1, float& p2) {
  const float* xp = x + idx * 3;
  float X = xp[0], Y = xp[1], Z = xp[2];
  float q0 = rt[0] * X + rt[1] * Y + rt[2] * Z + rt[9];
  float q1 = rt[3] * X + rt[4] * Y + rt[5] * Z + rt[10];
  float q2 = rt[6] * X + rt[7] * Y + rt[8] * Z + rt[11];
  const float4* yp = (const float4*)(y + idx * 16);
  float4 r0 = yp[0], r1 = yp[1], r2 = yp[2];
  p0 = r0.x * q0 + r0.y * q1 + r0.z * q2 + r0.w;
  p1 = r1.x * q0 + r1.y * q1 + r1.z * q2 + r1.w;
  p2 = r2.x * q0 + r2.y * q1 + r2.z * q2 + r2.w;
}

// ------------------------------------------------- fused transform + moments
// q = [x|1] @ [R;t]^T via V_WMMA_F32_16X16X4_F32 (16 points / WMMA, K=4),
// A operand loaded straight from memory in the ISA A-layout (no shuffles),
// D de-striped through a per-wave LDS tile, then per-lane y-matvec.
__global__ __launch_bounds__(256) void k_transform(
    const float* __restrict__ x, const float* __restrict__ y,
    const float* __restrict__ Rt, float* __restrict__ pbuf,
    double* __restrict__ mom, int store_p) {
  __shared__ float qtile[8][32][4];   // [wave][point-in-tile][coord(+pad)]
  __shared__ double bsum[9];
  const int tid  = threadIdx.x;
  const int lane = tid & 31;
  const int wv   = tid >> 5;
  const int j    = lane & 15;
  const bool hi  = lane >= 16;
  const int b    = blockIdx.x / BPB1;
  const int blk  = blockIdx.x % BPB1;
  if (tid < 9) bsum[tid] = 0.0;
  __syncthreads();

  float rt[12];
#pragma unroll
  for (int i = 0; i < 12; ++i) rt[i] = Rt[i];

  // B operand (4x16): B[k][j] = R[j][k] for k<3, B[3][j] = t[j].
  // 32-bit B layout assumed mirror of A: VGPR0 = rows K0|K2, VGPR1 = rows K1|K3.
  v2f bmat;
  {
    float b0 = 0.f, b1 = 0.f;
    if (j < 3) {
      if (!hi) { b0 = rt[j * 3 + 0]; b1 = rt[j * 3 + 1]; }
      else     { b0 = rt[j * 3 + 2]; b1 = rt[9 + j]; }
    }
    bmat.x = b0; bmat.y = b1;
  }

  const size_t base = (size_t)b * MPTS;
  float s1x = 0, s1y = 0, s1z = 0;
  float sxx = 0, sxy = 0, sxz = 0, syy = 0, syz = 0, szz = 0;

  const int tilesPerBatch = MPTS / 32;  // 15625 exact
  for (int tile = blk * 8 + wv; tile < tilesPerBatch; tile += BPB1 * 8) {
    const int m0 = tile * 32;
    // WMMA tile 0: points m0..m0+15   (A layout: lanes0-15 K0/K1, lanes16-31 K2/K3)
    v8f acc0 = {};
    {
      const float* xp = x + (base + m0 + j) * 3;
      v2f a;
      if (!hi) { a.x = xp[0]; a.y = xp[1]; }
      else     { a.x = xp[2]; a.y = 1.0f; }
      acc0 = __builtin_amdgcn_wmma_f32_16x16x4_f32(false, a, false, bmat,
                                                   (short)0, acc0, false, false);
    }
#pragma unroll
    for (int r = 0; r < 8; ++r)
      if (j < 3) qtile[wv][(hi ? 8 : 0) + r][j] = acc0[r];
    // WMMA tile 1: points m0+16..m0+31
    v8f acc1 = {};
    {
      const float* xp = x + (base + m0 + 16 + j) * 3;
      v2f a;
      if (!hi) { a.x = xp[0]; a.y = xp[1]; }
      else     { a.x = xp[2]; a.y = 1.0f; }
      acc1 = __builtin_amdgcn_wmma_f32_16x16x4_f32(false, a, false, bmat,
                                                   (short)0, acc1, false, false);
    }
#pragma unroll
    for (int r = 0; r < 8; ++r)
      if (j < 3) qtile[wv][16 + (hi ? 8 : 0) + r][j] = acc1[r];
    // cross-lane LDS visibility within the wave (DS is in-order per wave;
    // the barrier stops the compiler from hoisting the reads)
    asm volatile("s_wait_dscnt 0" ::: "memory");
    const int m = m0 + lane;
    float q0 = qtile[wv][lane][0];
    float q1 = qtile[wv][lane][1];
    float q2 = qtile[wv][lane][2];
    asm volatile("" ::: "memory");  // reads of this iter before stores of next
    const float4* yp = (const float4*)(y + (base + m) * 16);
    float4 r0 = yp[0], r1 = yp[1], r2 = yp[2];
    float p0 = r0.x * q0 + r0.y * q1 + r0.z * q2 + r0.w;
    float p1 = r1.x * q0 + r1.y * q1 + r1.z * q2 + r1.w;
    float p2 = r2.x * q0 + r2.y * q1 + r2.z * q2 + r2.w;
    if (store_p) {
      float* pp = pbuf + (base + m) * 3;
      pp[0] = p0; pp[1] = p1; pp[2] = p2;
    }
    s1x += p0; s1y += p1; s1z += p2;
    sxx += p0 * p0; sxy += p0 * p1; sxz += p0 * p2;
    syy += p1 * p1; syz += p1 * p2; szz += p2 * p2;
  }
  atomicAdd(&bsum[0], (double)s1x); atomicAdd(&bsum[1], (double)s1y);
  atomicAdd(&bsum[2], (double)s1z); atomicAdd(&bsum[3], (double)sxx);
  atomicAdd(&bsum[4], (double)sxy); atomicAdd(&bsum[5], (double)sxz);
  atomicAdd(&bsum[6], (double)syy); atomicAdd(&bsum[7], (double)syz);
  atomicAdd(&bsum[8], (double)szz);
  __syncthreads();
  if (tid < 9) atomicAdd(&mom[b * 9 + tid], bsum[tid]);
}

// ----------------------------------------- 3x3 covariance eigensolve (Jacobi)
__global__ void k_eigen(const double* __restrict__ mom, float* __restrict__ nd) {
  int b = threadIdx.x;
  if (b >= BS || blockIdx.x > 0) return;
  const double* s = mom + b * 9;
  const double Mn = (double)MPTS;
  double c0 = s[0] / Mn, c1 = s[1] / Mn, c2 = s[2] / Mn;
  double A[3][3];
  A[0][0] = s[3] - Mn * c0 * c0; A[0][1] = s[4] - Mn * c0 * c1;
  A[0][2] = s[5] - Mn * c0 * c2; A[1][1] = s[6] - Mn * c1 * c1;
  A[1][2] = s[7] - Mn * c1 * c2; A[2][2] = s[8] - Mn * c2 * c2;
  A[1][0] = A[0][1]; A[2][0] = A[0][2]; A[2][1] = A[1][2];
  double V[3][3] = {{1, 0, 0}, {0, 1, 0}, {0, 0, 1}};
  const int PP[3] = {0, 0, 1}, QQ[3] = {1, 2, 2};
  for (int sweep = 0; sweep < 30; ++sweep)
    for (int r = 0; r < 3; ++r) {
      int p = PP[r], q = QQ[r];
      double apq = A[p][q];
      if (fabs(apq) < 1e-300) continue;
      double th = (A[q][q] - A[p][p]) / (2.0 * apq);
      double t  = (th >= 0.0 ? 1.0 : -1.0) / (fabs(th) + sqrt(th * th + 1.0));
      double cc = 1.0 / sqrt(t * t + 1.0), ss = t * cc;
      for (int k = 0; k < 3; ++k) {
        double akp = A[k][p], akq = A[k][q];
        A[k][p] = cc * akp - ss * akq; A[k][q] = ss * akp + cc * akq;
      }
      for (int k = 0; k < 3; ++k) {
        double apk = A[p][k], aqk = A[q][k];
        A[p][k] = cc * apk - ss * aqk; A[q][k] = ss * apk + cc * aqk;
      }
      for (int k = 0; k < 3; ++k) {
        double vkp = V[k][p], vkq = V[k][q];
        V[k][p] = cc * vkp - ss * vkq; V[k][q] = ss * vkp + cc * vkq;
      }
    }
  int mi = 0;
  if (A[1][1] < A[mi][mi]) mi = 1;
  if (A[2][2] < A[mi][mi]) mi = 2;
  double n0 = V[0][mi], n1 = V[1][mi], n2 = V[2][mi];
  double nn = sqrt(n0 * n0 + n1 * n1 + n2 * n2);
  n0 /= nn; n1 /= nn; n2 /= nn;
  double dd = -(c0 * n0 + c1 * n1 + c2 * n2);
  nd[b * 4 + 0] = (float)n0; nd[b * 4 + 1] = (float)n1;
  nd[b * 4 + 2] = (float)n2; nd[b * 4 + 3] = (float)dd;
}

// residual l for one point
__device__ __forceinline__ unsigned int l_bits(
    const float* x, const float* y, const float* rt, const float* pbuf,
    int use_p, size_t idx, float& lval, float n0, float n1, float n2, float dd) {
  float p0, p1, p2;
  if (use_p) {
    const float* pp = pbuf + idx * 3;
    p0 = pp[0]; p1 = pp[1]; p2 = pp[2];
  } else {
    point_p(x, y, rt, idx, p0, p1, p2);
  }
  float xn = p0 * n0 + p1 * n1 + p2 * n2 + dd;
  lval = xn * xn;
  return __float_as_uint(lval);
}

// ------------------------------- level-1 histogram: top 16 bits, 256KB LDS
__global__ __launch_bounds__(1024) void k_hist1(
    const float* __restrict__ x, const float* __restrict__ y,
    const float* __restrict__ Rt, const float* __restrict__ pbuf, int use_p,
    const float* __restrict__ nd, unsigned int* __restrict__ hist) {
  extern __shared__ unsigned int lh[];   // HBINS u32 = 256 KB (CDNA5 LDS)
  const int b = blockIdx.x / BPB2, blk = blockIdx.x % BPB2;
  for (int i = threadIdx.x; i < HBINS; i += blockDim.x) lh[i] = 0u;
  __syncthreads();
  float n0 = nd[b * 4], n1 = nd[b * 4 + 1], n2 = nd[b * 4 + 2], dd = nd[b * 4 + 3];
  float rt[12];
#pragma unroll
  for (int i = 0; i < 12; ++i) rt[i] = Rt[i];
  const size_t base = (size_t)b * MPTS;
  for (int m = blk * blockDim.x + threadIdx.x; m < MPTS; m += BPB2 * blockDim.x) {
    float l;
    unsigned bits = l_bits(x, y, rt, pbuf, use_p, base + m, l, n0, n1, n2, dd);
    atomicAdd(&lh[bits >> 16], 1u);
  }
  __syncthreads();
  unsigned int* gh = hist + (size_t)b * HBINS;
  for (int i = threadIdx.x; i < HBINS; i += blockDim.x) {
    unsigned c = lh[i];
    if (c) atomicAdd(&gh[i], c);
  }
}

// find pivot hi-bin: first bin where cumulative count reaches KEEP
__global__ __launch_bounds__(1024) void k_scan1(const unsigned int* __restrict__ hist,
                                                unsigned int* __restrict__ sel1) {
  __shared__ unsigned int part[1024];
  const int b = blockIdx.x;
  const unsigned int* h = hist + (size_t)b * HBINS;
  unsigned s = 0;
  const int c0 = threadIdx.x * 64;
  for (int i = 0; i < 64; ++i) s += h[c0 + i];
  part[threadIdx.x] = s;
  __syncthreads();
  if (threadIdx.x == 0) {
    unsigned long long cum = 0;
    int ci = 0;
    for (; ci < 1023; ++ci) {
      if (cum + part[ci] >= (unsigned long long)KEEP) break;
      cum += part[ci];
    }
    unsigned bin = ci * 64;
    for (;; ++bin) {
      unsigned c = h[bin];
      if (cum + c >= (unsigned long long)KEEP) break;
      cum += c;
    }
    sel1[b * 2 + 0] = bin;
    sel1[b * 2 + 1] = (unsigned)cum;   // count strictly below pivot bin
  }
}

// ------------------------------- level-2 histogram: low 16 bits of pivot bin
__global__ __launch_bounds__(1024) void k_hist2(
    const float* __restrict__ x, const float* __restrict__ y,
    const float* __restrict__ Rt, const float* __restrict__ pbuf, int use_p,
    const float* __restrict__ nd, const unsigned int* __restrict__ sel1,
    unsigned int* __restrict__ hist) {
  extern __shared__ unsigned int lh[];
  const int b = blockIdx.x / BPB2, blk = blockIdx.x % BPB2;
  for (int i = threadIdx.x; i < HBINS; i += blockDim.x) lh[i] = 0u;
  __syncthreads();
  const unsigned pivot = sel1[b * 2];
  float n0 = nd[b * 4], n1 = nd[b * 4 + 1], n2 = nd[b * 4 + 2], dd = nd[b * 4 + 3];
  float rt[12];
#pragma unroll
  for (int i = 0; i < 12; ++i) rt[i] = Rt[i];
  const size_t base = (size_t)b * MPTS;
  for (int m = blk * blockDim.x + threadIdx.x; m < MPTS; m += BPB2 * blockDim.x) {
    float l;
    unsigned bits = l_bits(x, y, rt, pbuf, use_p, base + m, l, n0, n1, n2, dd);
    if ((bits >> 16) == pivot) atomicAdd(&lh[bits & 0xFFFFu], 1u);
  }
  __syncthreads();
  unsigned int* gh = hist + (size_t)b * HBINS;
  for (int i = threadIdx.x; i < HBINS; i += blockDim.x) {
    unsigned c = lh[i];
    if (c) atomicAdd(&gh[i], c);
  }
}

// exact threshold T (KEEP-th smallest) + count strictly below it
__global__ __launch_bounds__(1024) void k_scan2(const unsigned int* __restrict__ hist,
                                                const unsigned int* __restrict__ sel1,
                                                unsigned int* __restrict__ sel2) {
  __shared__ unsigned int part[1024];
  const int b = blockIdx.x;
  const unsigned int* h = hist + (size_t)b * HBINS;
  unsigned s = 0;
  const int c0 = threadIdx.x * 64;
  for (int i = 0; i < 64; ++i) s += h[c0 + i];
  part[threadIdx.x] = s;
  __syncthreads();
  if (threadIdx.x == 0) {
    unsigned long long cum = sel1[b * 2 + 1];
    int ci = 0;
    for (; ci < 1023; ++ci) {
      if (cum + part[ci] >= (unsigned long long)KEEP) break;
      cum += part[ci];
    }
    unsigned bin = ci * 64;
    for (;; ++bin) {
      unsigned c = h[bin];
      if (cum + c >= (unsigned long long)KEEP) break;
      cum += c;
    }
    sel2[b * 2 + 0] = (sel1[b * 2] << 16) | bin;  // bit pattern of threshold T
    sel2[b * 2 + 1] = (unsigned)cum;              // count of l strictly < T
  }
}

// ---------------------------------------- sum of residuals strictly below T
__global__ __launch_bounds__(512) void k_sum(
    const float* __restrict__ x, const float* __restrict__ y,
    const float* __restrict__ Rt, const float* __restrict__ pbuf, int use_p,
    const float* __restrict__ nd, const unsigned int* __restrict__ sel2,
    double* __restrict__ sums) {
  __shared__ double bs;
  const int b = blockIdx.x / BPB3, blk = blockIdx.x % BPB3;
  if (threadIdx.x == 0) bs = 0.0;
  __syncthreads();
  const unsigned T = sel2[b * 2];
  float n0 = nd[b * 4], n1 = nd[b * 4 + 1], n2 = nd[b * 4 + 2], dd = nd[b * 4 + 3];
  float rt[12];
#pragma unroll
  for (int i = 0; i < 12; ++i) rt[i] = Rt[i];
  const size_t base = (size_t)b * MPTS;
  float partial = 0.f;
  for (int m = blk * blockDim.x + threadIdx.x; m < MPTS; m += BPB3 * blockDim.x) {
    float l;
    unsigned bits = l_bits(x, y, rt, pbuf, use_p, base + m, l, n0, n1, n2, dd);
    if (bits < T) partial += l;
  }
  atomicAdd(&bs, (double)partial);
  __syncthreads();
  if (threadIdx.x == 0) atomicAdd(&sums[b], bs);
}

__global__ void k_final(const double* __restrict__ sums,
                        const unsigned int* __restrict__ sel2,
                        float* __restrict__ out) {
  if (threadIdx.x != 0 || blockIdx.x != 0) return;
  double tot = 0.0;
  for (int b = 0; b < BS; ++b) {
    unsigned T = sel2[b * 2], cl = sel2[b * 2 + 1];
    double Tv = (double)__uint_as_float(T);
    tot += sums[b] + (double)(KEEP - (long long)cl) * Tv;
  }
  out[0] = (float)(tot / (8.0 * (double)KEEP));
}

// ---------------------------------------------------------------- launcher
extern "C" void kernel_launch(void* const* d_in, const int* in_sizes, int n_in,
                              void* d_out, int out_size, void* d_ws, size_t ws_size,
                              hipStream_t stream) {
  const float* x  = (const float*)d_in[0];
  const float* y  = (const float*)d_in[1];
  const float* w  = (const float*)d_in[2];
  const float* tr = (const float*)d_in[3];
  float* out = (float*)d_out;
  char* ws = (char*)d_ws;

  float*        Rt    = (float*)(ws + 0);                    // 12 floats
  double*       mom   = (double*)(ws + 64);                  // 8*9 doubles
  double*       sums  = (double*)(ws + 640);                 // 8 doubles
  unsigned int* sel1  = (unsigned int*)(ws + 704);           // 8*2 u32
  unsigned int* sel2  = (unsigned int*)(ws + 768);           // 8*2 u32
  float*        nd    = (float*)(ws + 832);                  // 8*4 floats
  unsigned int* hist1 = (unsigned int*)(ws + 1024);          // 8*65536 u32
  unsigned int* hist2 = (unsigned int*)(ws + 1024 + (size_t)BS * HBINS * 4);
  const size_t p_bytes = (size_t)BS * MPTS * 3 * sizeof(float);
  const int store_p = (ws_size >= P_OFF + p_bytes) ? 1 : 0;
  float* pbuf = store_p ? (float*)(ws + P_OFF) : nullptr;

  // zero mom/sums/sel/nd + both histograms: bytes [64, 1024 + 2*hist)
  const int zwords = (int)(((1024 + 2 * (size_t)BS * HBINS * 4) - 64) / 4);
  k_zero<<<(zwords + 255) / 256, 256, 0, stream>>>((unsigned int*)(ws + 64), zwords);
  k_rt<<<1, 1, 0, stream>>>(w, tr, Rt);
  k_transform<<<BS * BPB1, 256, 0, stream>>>(x, y, Rt, pbuf, mom, store_p);
  k_eigen<<<1, BS, 0, stream>>>(mom, nd);
  k_hist1<<<BS * BPB2, 1024, HBINS * sizeof(unsigned int), stream>>>(
      x, y, Rt, pbuf, store_p, nd, hist1);
  k_scan1<<<BS, 1024, 0, stream>>>(hist1, sel1);
  k_hist2<<<BS * BPB2, 1024, HBINS * sizeof(unsigned int), stream>>>(
      x, y, Rt, pbuf, store_p, nd, sel1, hist2);
  k_scan2<<<BS, 1024, 0, stream>>>(hist2, sel1, sel2);
  k_sum<<<BS * BPB3, 512, 0, stream>>>(x, y, Rt, pbuf, store_p, nd, sel2, sums);
  k_final<<<1, 1, 0, stream>>>(sums, sel2, out);
}